// HeinnXConv2D_41532333752734
// MI455X (gfx1250) — compile-verified
//
#include <hip/hip_runtime.h>

// ---------------------------------------------------------------------------
// HeinnXConv2D for MI455X (gfx1250): two fused Chebyshev axis passes.
// fp32 end-to-end using V_WMMA_F32_16X16X4_F32 (memory-bound at 23.3 TB/s,
// so full-precision f32 matrix ops are free relative to the HBM roofline).
// ---------------------------------------------------------------------------

typedef float v2f __attribute__((ext_vector_type(2)));
typedef float v8f __attribute__((ext_vector_type(8)));

#define C_CH   96
#define N_PTS  192
#define D1     17     // degree+1 (base coeffs)
#define D2     18     // degree+2 (integral coeffs)
#define KP     20     // padded coeff columns (zero-filled 17..19)
#define KKC    36     // concat base(17)+int(18)+pad(1)
#define LPB    8      // lines per workgroup (8 | 192 -> never crosses batch)
#define NTHREADS 256
#define NWAVES 8
#define NPAD   196    // LDS row pitch in dwords (kills 64-bank conflicts)

// ---- LDS layout (float offsets), phase-overlaid, total 75,152 f = 293.6 KB
#define OFF_CM     0        // [768][20]  coeffs            (ph1-3)  15360
#define OFF_CINT   15360    // [768][20]  integral coeffs   (ph2-3)  15360
#define OFF_XS     15360    // [128][196] input stage       (ph1)    25088 (overlays CINT)
#define OFF_OB     0        // [128][196] output stage      (ph4)    25088 (overlays CM/CINT)
#define OFF_MCAT   40448    // [768][36]  m_base||m_int     (ph3-4)  27648
#define OFF_TPT    68096    // [32][196]  T_pinv padded     (ph1)     6272
#define OFF_ST     74368    // [20][32]   S^T padded        (ph2)      640
#define OFF_TCATT  68096    // [36][196]  [T||T_sum]^T      (ph4)     7056 (overlays TPT/ST)
#define LDS_FLOATS 75152

static __device__ __forceinline__ v8f wmma4(v2f a, v2f b, v8f c) {
    return __builtin_amdgcn_wmma_f32_16x16x4_f32(false, a, false, b,
                                                 (short)0, c, false, false);
}

// Repack Wb [96][96][17] -> [17][96][96] and Wi [96][96][18] -> [18][96][96]
// so phase-3 B-operand loads are o-contiguous (64B per 16 lanes, L2-resident).
__global__ void repack_weights_k(const float* __restrict__ Wb,
                                 const float* __restrict__ Wi,
                                 float* __restrict__ WbT,
                                 float* __restrict__ WiT) {
    int idx = blockIdx.x * blockDim.x + threadIdx.x;
    const int nb = D1 * C_CH * C_CH;
    const int ni = D2 * C_CH * C_CH;
    if (idx < nb) {
        int o = idx % C_CH, c = (idx / C_CH) % C_CH, k = idx / (C_CH * C_CH);
        WbT[idx] = Wb[(c * C_CH + o) * D1 + k];
    }
    if (idx < ni) {
        int o = idx % C_CH, c = (idx / C_CH) % C_CH, l = idx / (C_CH * C_CH);
        WiT[idx] = Wi[(c * C_CH + o) * D2 + l];
    }
}

// One spectral axis pass over LPB lines per workgroup.
// Row flattening: r = c*8 + ll  (ll = line-in-block); same for output with o.
// Strides (elements): sC = channel, sN = spectral axis, sL = line.
__launch_bounds__(NTHREADS)
__global__ void axis_pass_k(const float* in, float* out,
                            const float* __restrict__ Tp,    // [17][192]
                            const float* __restrict__ Tm,    // [192][17]
                            const float* __restrict__ Tsum,  // [192][18]
                            const float* __restrict__ Smat,  // [18][17]
                            const float* __restrict__ WbT,   // [17][96][96]
                            const float* __restrict__ WiT,   // [18][96][96]
                            int sC, int sN, int sL) {
    extern __shared__ float lds[];
    const int tid  = threadIdx.x;
    const int wv   = tid >> 5;          // wave32
    const int lane = tid & 31;
    const int half = lane >> 4;         // 0: K={0,1} / M 0..7 rows of D ; 1: K={2,3}
    const int l16  = lane & 15;

    const int line0 = blockIdx.x * LPB;
    const int b     = line0 / N_PTS;
    const int r0    = line0 % N_PTS;
    const long planeSz = (long)C_CH * N_PTS * N_PTS;
    const float* inB = in  + (long)b * planeSz + (long)r0 * sL;
    float*      outB = out + (long)b * planeSz + (long)r0 * sL;

    // ---- init: stage T_pinv^pad, S^T, zero Mcat pad column --------------
    for (int i = tid; i < 32 * N_PTS; i += NTHREADS) {
        int k = i / N_PTS, n = i % N_PTS;
        lds[OFF_TPT + k * NPAD + n] = (k < D1) ? Tp[k * N_PTS + n] : 0.0f;
    }
    for (int i = tid; i < KP * 32; i += NTHREADS) {
        int k = i / 32, l = i % 32;
        lds[OFF_ST + k * 32 + l] = (k < D1 && l < D2) ? Smat[l * D1 + k] : 0.0f;
    }
    for (int i = tid; i < LPB * C_CH; i += NTHREADS)
        lds[OFF_MCAT + i * KKC + (KKC - 1)] = 0.0f;
    __syncthreads();

    // ---- phase 1: Cm[r][k] = sum_n X[r][n] * Tp[k][n]  (K=192) ----------
    for (int sb = 0; sb < 6; ++sb) {                 // 6 x 128 rows
        if (sN == 1) {                               // W-pass: n contiguous
            for (int i = tid; i < 128 * (N_PTS / 4); i += NTHREADS) {
                int rl = i / (N_PTS / 4), nq = i % (N_PTS / 4);
                int row = sb * 128 + rl;
                int c = row >> 3, ll = row & 7;
                float4 v = *(const float4*)(inB + (long)c * sC + ll * sL + nq * 4);
                *(float4*)&lds[OFF_XS + rl * NPAD + nq * 4] = v;
            }
        } else {                                     // H-pass: line stride 1
            for (int i = tid; i < 128 * N_PTS; i += NTHREADS) {
                int ll = i & 7, rest = i >> 3;
                int n = rest % N_PTS, cl = rest / N_PTS;
                int c = sb * 16 + cl;
                lds[OFF_XS + (cl * 8 + ll) * NPAD + n] =
                    inB[(long)c * sC + (long)n * sN + ll * sL];
            }
        }
        __syncthreads();
        for (int kt = 0; kt < 2; ++kt) {             // wave wv owns row-tile wv
            v8f acc = {};
            const int abase = OFF_XS  + (wv * 16 + l16) * NPAD + 2 * half;
            const int bbase = OFF_TPT + (kt * 16 + l16) * NPAD + 2 * half;
            for (int n0 = 0; n0 < N_PTS; n0 += 4) {
                v2f a  = *(const v2f*)&lds[abase + n0];
                v2f bb = *(const v2f*)&lds[bbase + n0];
                acc = wmma4(a, bb, acc);
            }
            int k = kt * 16 + l16;
            if (k < KP) {                            // k 17..19 are exact zeros (Tpt pad)
                int rb = sb * 128 + wv * 16 + 8 * half;
#pragma unroll
                for (int q = 0; q < 8; ++q)
                    lds[OFF_CM + (rb + q) * KP + k] = acc[q];
            }
        }
        __syncthreads();
    }

    // ---- phase 2: Cint[r][l] = sum_k Cm[r][k] * S[l][k]  (K=20 padded) --
    for (int job = wv; job < 96; job += NWAVES) {    // 48 row-tiles x 2 l-tiles
        int rt = job >> 1, lt = job & 1;
        v8f acc = {};
        for (int k0 = 0; k0 < KP; k0 += 4) {
            v2f a = *(const v2f*)&lds[OFF_CM + (rt * 16 + l16) * KP + k0 + 2 * half];
            v2f bb;
            bb.x = lds[OFF_ST + (k0 + 2 * half + 0) * 32 + lt * 16 + l16];
            bb.y = lds[OFF_ST + (k0 + 2 * half + 1) * 32 + lt * 16 + l16];
            acc = wmma4(a, bb, acc);
        }
        int l = lt * 16 + l16;
        if (l < KP) {
            int rb = rt * 16 + 8 * half;
#pragma unroll
            for (int q = 0; q < 8; ++q)
                lds[OFF_CINT + (rb + q) * KP + l] = acc[q];
        }
    }
    __syncthreads();

    // ---- phase 3: per-k channel mix, M=lines(8 valid), K=96(c), N=96(o) -
    // Mcat[(o*8+ll)][kc] with kc = k (base) or 17+l (integral)
    for (int job = wv; job < (D1 + D2) * 6; job += NWAVES) {
        int kc, ot, Aoff;
        const float* Bp;
        if (job < D1 * 6) {
            int k = job / 6; ot = job % 6;
            Bp = WbT + k * C_CH * C_CH; Aoff = OFF_CM + k;  kc = k;
        } else {
            int jj = job - D1 * 6; int l = jj / 6; ot = jj % 6;
            Bp = WiT + l * C_CH * C_CH; Aoff = OFF_CINT + l; kc = D1 + l;
        }
        v8f acc = {};
        for (int c0 = 0; c0 < C_CH; c0 += 4) {
            int ca = c0 + 2 * half;
            v2f a, bb;
            a.x  = lds[Aoff + ((ca + 0) * 8 + l16) * KP];   // rows 8..15: garbage, discarded
            a.y  = lds[Aoff + ((ca + 1) * 8 + l16) * KP];
            bb.x = Bp[(ca + 0) * C_CH + ot * 16 + l16];     // L2-resident repacked weights
            bb.y = Bp[(ca + 1) * C_CH + ot * 16 + l16];
            acc = wmma4(a, bb, acc);
        }
        if (half == 0) {                                    // only D rows 0..7 (= lines) valid
#pragma unroll
            for (int q = 0; q < 8; ++q)
                lds[OFF_MCAT + ((ot * 16 + l16) * 8 + q) * KKC + kc] = acc[q];
        }
    }
    __syncthreads();

    // ---- phase 4: synthesis  out = Mcat * [T || T_sum]^T  (K=36) --------
    for (int i = tid; i < KKC * N_PTS; i += NTHREADS) {
        int kk = i / N_PTS, n = i % N_PTS;
        float v = 0.0f;
        if (kk < D1)            v = Tm[n * D1 + kk];
        else if (kk < D1 + D2)  v = Tsum[n * D2 + (kk - D1)];
        lds[OFF_TCATT + kk * NPAD + n] = v;
    }
    __syncthreads();
    for (int sb = 0; sb < 6; ++sb) {                 // 128 output rows per sub-block
        for (int job = wv; job < 96; job += NWAVES) {
            int rt = job / 12, nt = job % 12;
            v8f acc = {};
            const int abase = OFF_MCAT + (sb * 128 + rt * 16 + l16) * KKC + 2 * half;
            const int bbase = OFF_TCATT + nt * 16 + l16;
            for (int k0 = 0; k0 < KKC; k0 += 4) {
                v2f a = *(const v2f*)&lds[abase + k0];
                v2f bb;
                bb.x = lds[bbase + (k0 + 2 * half + 0) * NPAD];
                bb.y = lds[bbase + (k0 + 2 * half + 1) * NPAD];
                acc = wmma4(a, bb, acc);
            }
#pragma unroll
            for (int q = 0; q < 8; ++q)
                lds[OFF_OB + (rt * 16 + 8 * half + q) * NPAD + nt * 16 + l16] = acc[q];
        }
        __syncthreads();
        if (sN == 1) {                               // coalesced 16B stores (W-pass)
            for (int i = tid; i < 128 * (N_PTS / 4); i += NTHREADS) {
                int rl = i / (N_PTS / 4), nq = i % (N_PTS / 4);
                int row = sb * 128 + rl;
                int o = row >> 3, ll = row & 7;
                float4 v = *(const float4*)&lds[OFF_OB + rl * NPAD + nq * 4];
                *(float4*)(outB + (long)o * sC + ll * sL + nq * 4) = v;
            }
        } else {                                     // line-fast stores (H-pass)
            for (int i = tid; i < 128 * N_PTS; i += NTHREADS) {
                int ll = i & 7, rest = i >> 3;
                int n = rest % N_PTS, ol = rest / N_PTS;
                int o = sb * 16 + ol;
                outB[(long)o * sC + (long)n * sN + ll * sL] =
                    lds[OFF_OB + (ol * 8 + ll) * NPAD + n];
            }
        }
        __syncthreads();
    }
}

extern "C" void kernel_launch(void* const* d_in, const int* in_sizes, int n_in,
                              void* d_out, int out_size, void* d_ws, size_t ws_size,
                              hipStream_t stream) {
    (void)in_sizes; (void)n_in; (void)out_size; (void)ws_size;
    const float* x    = (const float*)d_in[0];
    const float* Wbw  = (const float*)d_in[1];
    const float* Wbh  = (const float*)d_in[2];
    const float* Wiw  = (const float*)d_in[3];
    const float* Wih  = (const float*)d_in[4];
    const float* S    = (const float*)d_in[5];
    const float* Tp   = (const float*)d_in[6];
    const float* Tm   = (const float*)d_in[7];
    const float* Tsum = (const float*)d_in[8];
    float* out = (float*)d_out;

    // workspace: repacked weights (2.46 MB total)
    float* WbTw = (float*)d_ws;
    float* WiTw = WbTw + D1 * C_CH * C_CH;
    float* WbTh = WiTw + D2 * C_CH * C_CH;
    float* WiTh = WbTh + D1 * C_CH * C_CH;

    const int nrep = D2 * C_CH * C_CH;
    repack_weights_k<<<(nrep + 255) / 256, 256, 0, stream>>>(Wbw, Wiw, WbTw, WiTw);
    repack_weights_k<<<(nrep + 255) / 256, 256, 0, stream>>>(Wbh, Wih, WbTh, WiTh);

    const size_t shmem = LDS_FLOATS * sizeof(float);   // ~294 KB (< 320 KB/WGP)
    const int nblocks = (16 * N_PTS) / LPB;            // 3072 lines / 8 = 384

    // W-axis pass: line = (b,h); strides c:H*W, n(=w):1, line(=h):W
    axis_pass_k<<<nblocks, NTHREADS, shmem, stream>>>(
        x, out, Tp, Tm, Tsum, S, WbTw, WiTw, N_PTS * N_PTS, 1, N_PTS);
    // H-axis pass, in-place: line = (b,w); strides c:H*W, n(=h):W, line(=w):1
    // Safe: each WG fully reads its 8 lines (phase 1) before writing them (phase 4).
    axis_pass_k<<<nblocks, NTHREADS, shmem, stream>>>(
        out, out, Tp, Tm, Tsum, S, WbTh, WiTh, N_PTS * N_PTS, N_PTS, 1);
}